// SAGE_13743895347603
// MI455X (gfx1250) — compile-verified
//
#include <hip/hip_runtime.h>

typedef float v2f __attribute__((ext_vector_type(2)));
typedef float v8f __attribute__((ext_vector_type(8)));

#define D_H 128
#define NEG_SLOPE 0.1f

// ---------------------------------------------------------------------------
// Utility kernels
// ---------------------------------------------------------------------------
__global__ void zero_f32(float* __restrict__ p, int n) {
    int t = blockIdx.x * blockDim.x + threadIdx.x;
    if (t < n) p[t] = 0.0f;
}

__global__ void count_deg(const int* __restrict__ dst, float* __restrict__ cnt, int e) {
    int t = blockIdx.x * blockDim.x + threadIdx.x;
    if (t < e) atomicAdd(&cnt[dst[t]], 1.0f);
}

__global__ void inv_deg(float* __restrict__ cnt, int n) {
    int t = blockIdx.x * blockDim.x + threadIdx.x;
    if (t < n) cnt[t] = 1.0f / fmaxf(cnt[t], 1.0f);
}

// One wave (32 lanes) per edge; each lane moves 4 consecutive floats of the
// 128-float feature row: coalesced 512B row read, f32 atomics into agg[dst].
__global__ __launch_bounds__(256) void scatter_add(
    const float* __restrict__ x, const int* __restrict__ src,
    const int* __restrict__ dst, float* __restrict__ agg, int e) {
    long long t = (long long)blockIdx.x * blockDim.x + threadIdx.x;
    int edge = (int)(t >> 5);
    int lane = (int)(t & 31);
    if (edge >= e) return;
    int s = src[edge];
    int d = dst[edge];
    const float4 v = *(const float4*)(x + (size_t)s * D_H + lane * 4);
    float* o = agg + (size_t)d * D_H + lane * 4;
    atomicAdd(o + 0, v.x);
    atomicAdd(o + 1, v.y);
    atomicAdd(o + 2, v.z);
    atomicAdd(o + 3, v.w);
}

// ---------------------------------------------------------------------------
// Fused dual-GEMM via V_WMMA_F32_16X16X4_F32:
//   out[n, j] = act( (agg[n,:] * inv[n]) @ Wa[:, j] + x[n,:] @ Wx[:, j] + bias[j] )
// Wa/Wx are [128, dout] row-major. One wave computes one 16x16 output tile.
// A-layout (f32 16x4): lane m (0-15) holds K={0,1}, lane m+16 holds K={2,3}.
// C/D layout: vgpr v on lane l -> row (v + 8*(l>=16)), col (l%16).
// ---------------------------------------------------------------------------
template <bool HAS_AGG, bool LRELU>
__global__ __launch_bounds__(256) void gemm_wmma(
    const float* __restrict__ agg, const float* __restrict__ inv,
    const float* __restrict__ x, const float* __restrict__ Wa,
    const float* __restrict__ Wx, const float* __restrict__ bias,
    float* __restrict__ out, int n, int dout) {
    const int col_tiles = dout >> 4;
    const int total = (n >> 4) * col_tiles;
    const int wid = threadIdx.x >> 5;
    const int tile = blockIdx.x * 8 + wid;
    if (tile >= total) return;  // wave-uniform: EXEC stays all-ones for WMMA

    const int lane = threadIdx.x & 31;
    const int idx = lane & 15;   // col within tile; also A-row for loads
    const int hi = lane >> 4;    // selects K pair {0,1} vs {2,3}
    const int rt = tile / col_tiles;
    const int ct = tile - rt * col_tiles;
    const int row0 = rt << 4;
    const int col = (ct << 4) + idx;

    const float* xrow = x + (size_t)(row0 + idx) * D_H;
    const float* arow = HAS_AGG ? (agg + (size_t)(row0 + idx) * D_H) : nullptr;
    const float invc = HAS_AGG ? inv[row0 + idx] : 0.0f;

    v8f acc = {};
#pragma unroll 8
    for (int k = 0; k < D_H; k += 4) {
        const int ka = k + (hi << 1);
        // x @ Wx contribution
        v2f a = *(const v2f*)(xrow + ka);
        v2f b;
        b.x = Wx[(size_t)ka * dout + col];
        b.y = Wx[(size_t)(ka + 1) * dout + col];
        acc = __builtin_amdgcn_wmma_f32_16x16x4_f32(
            false, a, false, b, (short)0, acc, false, false);
        if (HAS_AGG) {
            // (agg * inv) @ Wa contribution
            v2f a2 = *(const v2f*)(arow + ka);
            a2.x *= invc;
            a2.y *= invc;
            v2f b2;
            b2.x = Wa[(size_t)ka * dout + col];
            b2.y = Wa[(size_t)(ka + 1) * dout + col];
            acc = __builtin_amdgcn_wmma_f32_16x16x4_f32(
                false, a2, false, b2, (short)0, acc, false, false);
        }
    }

    const float bv = bias[col];
#pragma unroll
    for (int v = 0; v < 8; ++v) {
        const int r = row0 + v + (hi << 3);
        float val = acc[v] + bv;
        if (LRELU) val = val > 0.0f ? val : NEG_SLOPE * val;
        out[(size_t)r * dout + col] = val;
    }
}

// ---------------------------------------------------------------------------
// Host-side orchestration
// ---------------------------------------------------------------------------
extern "C" void kernel_launch(void* const* d_in, const int* in_sizes, int n_in,
                              void* d_out, int out_size, void* d_ws, size_t ws_size,
                              hipStream_t stream) {
    const float* x    = (const float*)d_in[0];
    const int*   ei   = (const int*)d_in[1];
    const float* wl   = (const float*)d_in[2];   // [3,128,128]
    const float* wr   = (const float*)d_in[3];   // [3,128,128]
    const float* bc   = (const float*)d_in[4];   // [3,128]
    const float* wlin = (const float*)d_in[5];   // [2,128,128]
    const float* blin = (const float*)d_in[6];   // [2,128]
    const float* wout = (const float*)d_in[7];   // [128,64]
    const float* bout = (const float*)d_in[8];   // [64]
    float* out = (float*)d_out;

    const int N = in_sizes[0] / D_H;    // 50000
    const int E = in_sizes[1] / 2;      // 800000
    const int DOUT = 64;

    float* bufA = (float*)d_ws;                  // features ping
    float* bufB = bufA + (size_t)N * D_H;        // features pong
    float* aggb = bufB + (size_t)N * D_H;        // neighbor aggregate
    float* cnt  = aggb + (size_t)N * D_H;        // degree -> inverse degree

    const int* src = ei;
    const int* dst = ei + E;

    // Degree counts (constant across layers)
    zero_f32<<<(N + 255) / 256, 256, 0, stream>>>(cnt, N);
    count_deg<<<(E + 255) / 256, 256, 0, stream>>>(dst, cnt, E);
    inv_deg<<<(N + 255) / 256, 256, 0, stream>>>(cnt, N);

    const int nd = N * D_H;
    const long long scat_threads = (long long)E * 32;
    const int scat_blocks = (int)((scat_threads + 255) / 256);
    const int tiles_h = (N >> 4) * (D_H >> 4);   // dout = 128
    const int tiles_o = (N >> 4) * (DOUT >> 4);  // dout = 64
    const int gblk_h = (tiles_h + 7) / 8;
    const int gblk_o = (tiles_o + 7) / 8;

    const float* cur = x;  // layer-0 input read directly (inputs never mutated)
    for (int l = 0; l < 3; ++l) {
        zero_f32<<<(nd + 255) / 256, 256, 0, stream>>>(aggb, nd);
        scatter_add<<<scat_blocks, 256, 0, stream>>>(cur, src, dst, aggb, E);

        const float* Wl = wl + (size_t)l * D_H * D_H;
        const float* Wr = wr + (size_t)l * D_H * D_H;
        const float* Bc = bc + (size_t)l * D_H;

        if (l < 2) {
            // sage conv + leaky relu : cur -> bufB
            gemm_wmma<true, true><<<gblk_h, 256, 0, stream>>>(
                aggb, cnt, cur, Wl, Wr, Bc, bufB, N, D_H);
            // hidden linear + leaky relu : bufB -> bufA
            gemm_wmma<false, true><<<gblk_h, 256, 0, stream>>>(
                nullptr, nullptr, bufB, nullptr,
                wlin + (size_t)l * D_H * D_H, blin + (size_t)l * D_H,
                bufA, N, D_H);
            cur = bufA;
        } else {
            // final sage conv (no activation) : cur -> bufB
            gemm_wmma<true, false><<<gblk_h, 256, 0, stream>>>(
                aggb, cnt, cur, Wl, Wr, Bc, bufB, N, D_H);
            // output projection 128 -> 64 : bufB -> d_out
            gemm_wmma<false, false><<<gblk_o, 256, 0, stream>>>(
                nullptr, nullptr, bufB, nullptr, wout, bout, out, N, DOUT);
        }
    }
}